// Order1Uncorrelated_77429670412927
// MI455X (gfx1250) — compile-verified
//
#include <hip/hip_runtime.h>
#include <hip/hip_bf16.h>

typedef __attribute__((ext_vector_type(2))) float v2f;
typedef __attribute__((ext_vector_type(8))) float v8f;
typedef __attribute__((ext_vector_type(4))) int   v4i;

typedef __attribute__((address_space(1))) v4i* gv4i_p;   // global int4*
typedef __attribute__((address_space(3))) v4i* lv4i_p;   // LDS int4*

#define B_   128
#define IN_  256
#define H_   128
#define OUT_ 32
#define K_   512          // 2*IN_
#define N_   4096         // H_*OUT_
#define NT_  (N_ / 16)    // 256 n-tiles
#define KBSTRIDE_ (NT_ * 64)   // floats between consecutive k-blocks = 16384

// workspace layout in floats
#define XEXT_OFF 0
#define MEXT_OFF (B_ * K_)                       // 65536
#define C0_OFF   (MEXT_OFF + K_ * (size_t)N_)    // 2162688
#define PW_OFF   (C0_OFF + N_)                   // 2166784
#define PWE_OFF  (PW_OFF + B_ * OUT_ * 8)        // 2199552

// fragment-packed index for Mext: block (k/4, n/16) of 64 floats, [n%16][k%4]
__device__ __forceinline__ size_t pkidx(int k, int n) {
    return ((size_t)(k >> 2) * NT_ + (n >> 4)) * 64 + (size_t)(n & 15) * 4 + (k & 3);
}

// ---------------- prep: Xext[b][k] = k<IN ? x : x^2 ----------------
__global__ void prep_x(const float* __restrict__ x, float* __restrict__ Xext) {
    int idx = blockIdx.x * blockDim.x + threadIdx.x;   // B_*K_ = 65536
    if (idx >= B_ * K_) return;
    int b = idx / K_, k = idx % K_;
    float v = (k < IN_) ? x[b * IN_ + k] : x[b * IN_ + (k - IN_)];
    Xext[idx] = (k < IN_) ? v : v * v;
}

// -- prep: Mext rows [0,IN) = -2*sp^2*chi ; rows [IN,2IN) = sp^2 (packed) --
__global__ void prep_m(const float* __restrict__ sigma_psi,
                       const float* __restrict__ chi,
                       float* __restrict__ Mext) {
    int idx = blockIdx.x * blockDim.x + threadIdx.x;   // IN_*N_ = 1048576
    if (idx >= IN_ * N_) return;
    int i = idx / N_, n = idx % N_;
    int h = n / OUT_;
    float sp = sigma_psi[idx];
    float A  = sp * sp;
    float ch = chi[i * H_ + h];
    Mext[pkidx(i, n)]        = -2.0f * A * ch;
    Mext[pkidx(IN_ + i, n)]  = A;
}

// ---------------- prep: c0[n] = sum_i sp^2 * chi^2 ----------------
__global__ void prep_c0(const float* __restrict__ sigma_psi,
                        const float* __restrict__ chi,
                        float* __restrict__ c0) {
    int n = blockIdx.x * blockDim.x + threadIdx.x;     // N_ = 4096
    if (n >= N_) return;
    int h = n / OUT_;
    float acc = 0.0f;
    for (int i = 0; i < IN_; ++i) {
        float sp = sigma_psi[(size_t)i * N_ + n];
        float ch = chi[i * H_ + h];
        acc += sp * sp * ch * ch;
    }
    c0[n] = acc;
}

// -------- main: WMMA GEMM + fused w-epilogue + in-WG reduction --------
// grid.x = 8(btile) * 2(otile) * 8(hchunk) = 128 ; block = 256 (8 waves)
__global__ void __launch_bounds__(256)
gemm_fused(const float* __restrict__ Xext, const float* __restrict__ Mext,
           const float* __restrict__ c0,   const float* __restrict__ sigma_eps,
           const float* __restrict__ eta,
           float* __restrict__ pw, float* __restrict__ pwe) {
    __shared__ float lA[16 * K_];          // 32 KB: 16 rows of Xext
    __shared__ float redW[8 * 16 * 16];    // 8 KB
    __shared__ float redE[8 * 16 * 16];    // 8 KB

    const int bt = blockIdx.x & 7;
    const int ot = (blockIdx.x >> 3) & 1;
    const int hc = blockIdx.x >> 4;
    const int tid   = threadIdx.x;
    const int wave  = tid >> 5;
    const int lane  = tid & 31;
    const int lrow  = lane & 15;           // N index within 16-tile (and A-row M)
    const int khalf = lane >> 4;           // K sub-offset selector {0,1}
    const int b0 = bt * 16;

    // stage A tile (16 x 512 f32) into LDS
#if __has_builtin(__builtin_amdgcn_global_load_async_to_lds_b128)
    for (int v = tid; v < (16 * K_) / 4; v += 256) {
        const float4* g = ((const float4*)(Xext + (size_t)b0 * K_)) + v;
        float4*       l = ((float4*)lA) + v;
        __builtin_amdgcn_global_load_async_to_lds_b128(
            (gv4i_p)(uintptr_t)g,
            (lv4i_p)(uint32_t)(uintptr_t)l,
            0, 0);
    }
    asm volatile("s_wait_asynccnt 0x0" ::: "memory");
#else
    for (int v = tid; v < (16 * K_) / 4; v += 256)
        ((float4*)lA)[v] = ((const float4*)(Xext + (size_t)b0 * K_))[v];
#endif
    __syncthreads();

    // two h values per wave: h0, h0+1 ; ntile = h*2 + ot
    const int h0 = hc * 16 + wave * 2;
    const float* Ap  = lA + lrow * K_ + khalf * 2;
    const float* Bp0 = Mext + (size_t)(h0 * 2 + ot) * 64 + lrow * 4 + khalf * 2;
    const float* Bp1 = Bp0 + 2 * 64;   // next h -> ntile + 2

    v8f acc0 = {}, acc1 = {};
    for (int k = 0; k < K_; k += 4) {
        const size_t kb = (size_t)(k >> 2) * KBSTRIDE_;
        v2f a   = *(const v2f*)(Ap + k);
        v2f bf0 = *(const v2f*)(Bp0 + kb);
        v2f bf1 = *(const v2f*)(Bp1 + kb);
        __builtin_prefetch(Bp0 + kb + KBSTRIDE_, 0, 0);
        acc0 = __builtin_amdgcn_wmma_f32_16x16x4_f32(
                 false, a, false, bf0, (short)0, acc0, false, false);
        acc1 = __builtin_amdgcn_wmma_f32_16x16x4_f32(
                 false, a, false, bf1, (short)0, acc1, false, false);
    }

    // epilogue: s -> w = 1/(s + se^2); accumulate w, w*eta over the 2 h's
    float accw[8], accwe[8];
#pragma unroll
    for (int r = 0; r < 8; ++r) { accw[r] = 0.0f; accwe[r] = 0.0f; }

    v8f accs[2] = {acc0, acc1};
#pragma unroll
    for (int t = 0; t < 2; ++t) {
        const int h   = h0 + t;
        const int n0  = h * OUT_ + ot * 16;
        const int o   = ot * 16 + lrow;
        const float c0v = c0[n0 + lrow];
        const float se  = sigma_eps[h * OUT_ + o];
        const float ev  = eta[h * OUT_ + o];
        const float se2 = se * se;
#pragma unroll
        for (int r = 0; r < 8; ++r) {
            float s = accs[t][r] + c0v;
            float w = 1.0f / (s + se2);
            accw[r]  += w;
            accwe[r] += w * ev;
        }
    }

    // lane L reg r holds (b_local = r + 8*khalf, o_local = lrow)
#pragma unroll
    for (int r = 0; r < 8; ++r) {
        int bl = r + 8 * khalf;
        redW[(wave * 16 + bl) * 16 + lrow] = accw[r];
        redE[(wave * 16 + bl) * 16 + lrow] = accwe[r];
    }
    __syncthreads();

    // fixed-order cross-wave reduction: thread = (b_local, o_local)
    const int bl = tid >> 4, ol = tid & 15;
    float sw = 0.0f, swe = 0.0f;
#pragma unroll
    for (int wv = 0; wv < 8; ++wv) {
        sw  += redW[(wv * 16 + bl) * 16 + ol];
        swe += redE[(wv * 16 + bl) * 16 + ol];
    }
    const int b = b0 + bl;
    const int o = ot * 16 + ol;
    pw [((size_t)b * OUT_ + o) * 8 + hc] = sw;
    pwe[((size_t)b * OUT_ + o) * 8 + hc] = swe;
}

// ---------------- final: fold h-chunk partials, posterior ----------------
__global__ void finalize(const float* __restrict__ pw, const float* __restrict__ pwe,
                         const float* __restrict__ mu_phi,
                         const float* __restrict__ sigma_phi,
                         float* __restrict__ out) {
    int idx = blockIdx.x * blockDim.x + threadIdx.x;   // B_*OUT_ = 4096
    if (idx >= B_ * OUT_) return;
    int o = idx % OUT_;
    float sw = 0.0f, swe = 0.0f;
#pragma unroll
    for (int c = 0; c < 8; ++c) {
        sw  += pw [(size_t)idx * 8 + c];
        swe += pwe[(size_t)idx * 8 + c];
    }
    float sp  = sigma_phi[o];
    float sp2 = sp * sp;
    float denom = 1.0f + sp2 * sw;
    out[idx]             = (mu_phi[o] + sp2 * swe) / denom;  // post_mu
    out[B_ * OUT_ + idx] = sp2 / denom;                      // post_sigma2
}

extern "C" void kernel_launch(void* const* d_in, const int* in_sizes, int n_in,
                              void* d_out, int out_size, void* d_ws, size_t ws_size,
                              hipStream_t stream) {
    const float* input     = (const float*)d_in[0];
    const float* sigma_psi = (const float*)d_in[1];
    const float* chi       = (const float*)d_in[2];
    const float* sigma_eps = (const float*)d_in[3];
    const float* eta       = (const float*)d_in[4];
    const float* mu_phi    = (const float*)d_in[5];
    const float* sigma_phi = (const float*)d_in[6];
    float* out = (float*)d_out;
    float* ws  = (float*)d_ws;

    float* Xext = ws + XEXT_OFF;
    float* Mext = ws + MEXT_OFF;
    float* c0   = ws + C0_OFF;
    float* pw   = ws + PW_OFF;
    float* pwe  = ws + PWE_OFF;

    prep_x <<<(B_ * K_ + 255) / 256, 256, 0, stream>>>(input, Xext);
    prep_m <<<(IN_ * N_ + 255) / 256, 256, 0, stream>>>(sigma_psi, chi, Mext);
    prep_c0<<<(N_ + 255) / 256, 256, 0, stream>>>(sigma_psi, chi, c0);
    gemm_fused<<<128, 256, 0, stream>>>(Xext, Mext, c0, sigma_eps, eta, pw, pwe);
    finalize<<<(B_ * OUT_ + 255) / 256, 256, 0, stream>>>(pw, pwe, mu_phi, sigma_phi, out);
}